// GNN_att_64759516889769
// MI455X (gfx1250) — compile-verified
//
#include <hip/hip_runtime.h>
#include <math.h>

#define N_NODES 8064
#define D 128
#define DEG 32
#define NPS 1008   // nodes per subject
#define NSUBJ 8

typedef __attribute__((ext_vector_type(2))) float v2f;
typedef __attribute__((ext_vector_type(8))) float v8f;

__device__ __forceinline__ v8f wmma_f32(v2f a, v2f b, v8f c) {
  // V_WMMA_F32_16X16X4_F32 : D = A(16x4 f32) x B(4x16 f32) + C(16x16 f32)
  return __builtin_amdgcn_wmma_f32_16x16x4_f32(
      /*neg_a=*/false, a, /*neg_b=*/false, b,
      /*c_mod=*/(short)0, c, /*reuse_a=*/false, /*reuse_b=*/false);
}

// ---------------------------------------------------------------------------
// GIN aggregation: out[i,:] = x[i,:] + sum_{k<32} x[src[i*32+k],:]
// ---------------------------------------------------------------------------
__global__ void gin_agg_kernel(const float* __restrict__ x, const int* __restrict__ src,
                               float* __restrict__ out, int F) {
  __shared__ int s_src[DEG];
  int node = blockIdx.x;
  if (threadIdx.x < DEG) s_src[threadIdx.x] = src[node * DEG + threadIdx.x];
  __syncthreads();
  for (int f = threadIdx.x; f < F; f += blockDim.x) {
    float acc = x[(long)node * F + f];
#pragma unroll
    for (int k = 0; k < DEG; ++k) acc += x[(long)s_src[k] * F + f];
    out[(long)node * F + f] = acc;
  }
}

// ---------------------------------------------------------------------------
// WMMA fp32 GEMM: out[M x 128] = A[M x K] @ W[K x 128] + bias (+ tiled pe)
// One wave computes one 16x16 tile. K must be a multiple of 4 (84 or 128).
// ---------------------------------------------------------------------------
__global__ void wmma_gemm_kernel(const float* __restrict__ A, const float* __restrict__ W,
                                 const float* __restrict__ bias, const float* __restrict__ pe,
                                 float* __restrict__ out, int K) {
  int m0 = blockIdx.x * 16;
  int n0 = blockIdx.y * 16;
  int lane = threadIdx.x & 31;
  int half = lane >> 4;   // 0 or 1
  int l16  = lane & 15;

  v8f c = {};
  const float* arow = A + (long)(m0 + l16) * K;
  const v2f* arv = (const v2f*)arow;   // rows are 8B aligned (K even)
  for (int k = 0; k < K; k += 4) {
    int ka = k + 2 * half;             // even
    v2f a = arv[ka >> 1];              // A[m][ka], A[m][ka+1]
    v2f b;
    b.x = W[ka * D + n0 + l16];        // B[k][n] rows ka, ka+1
    b.y = W[(ka + 1) * D + n0 + l16];
    c = wmma_f32(a, b, c);
  }
  float bv = bias[n0 + l16];
#pragma unroll
  for (int r = 0; r < 8; ++r) {
    int row = m0 + r + 8 * half;
    float v = c[r] + bv;
    if (pe) v += pe[(row % NPS) * D + n0 + l16];
    out[(long)row * D + n0 + l16] = v;
  }
}

// ---------------------------------------------------------------------------
// Flash attention over all N nodes: ctx = softmax(h2 h2^T / sqrt(128)) h2
// One wave per 16-query tile, online softmax, fp32 WMMA throughout.
// ---------------------------------------------------------------------------
__global__ void flash_attn_kernel(const float* __restrict__ h2, float* __restrict__ ctx) {
  __shared__ float s_p[4 * 16 * 16];   // one 16x16 P slab per wave
  int warp = threadIdx.x >> 5;
  int lane = threadIdx.x & 31;
  int half = lane >> 4;
  int l16  = lane & 15;
  int m0   = (blockIdx.x * 4 + warp) * 16;
  float* pp = s_p + warp * 256;
  const v2f* h2v2 = (const v2f*)h2;

  // Q fragments for 32 k-steps (A-layout), kept in registers
  v2f qa[32];
#pragma unroll
  for (int kk = 0; kk < 32; ++kk)
    qa[kk] = h2v2[(m0 + l16) * 64 + 2 * kk + half];

  v8f acc[8];
#pragma unroll
  for (int t = 0; t < 8; ++t) acc[t] = (v8f){};
  float mrow[8], lrow[8];
#pragma unroll
  for (int r = 0; r < 8; ++r) { mrow[r] = -INFINITY; lrow[r] = 0.f; }

  const float scale = 0.08838834764831845f;  // 1/sqrt(128)

  for (int j0 = 0; j0 < N_NODES; j0 += 16) {
    // S = Q K^T : B[k][n] = h2[j0+n][k]  (same load pattern as Q fragments)
    v8f s = {};
#pragma unroll
    for (int kk = 0; kk < 32; ++kk) {
      v2f b = h2v2[(j0 + l16) * 64 + 2 * kk + half];
      s = wmma_f32(qa[kk], b, s);
    }
    // online softmax per row (rows r+8*half live across one 16-lane half)
    float corr[8];
#pragma unroll
    for (int r = 0; r < 8; ++r) {
      float v = s[r] * scale;
      float mx = v;
#pragma unroll
      for (int off = 8; off >= 1; off >>= 1)
        mx = fmaxf(mx, __shfl_xor(mx, off, 16));
      float mnew = fmaxf(mrow[r], mx);
      corr[r] = expf(mrow[r] - mnew);   // exp(-inf - finite) = 0 on first tile
      mrow[r] = mnew;
      float p = expf(v - mnew);
      s[r] = p;
      float sum = p;
#pragma unroll
      for (int off = 8; off >= 1; off >>= 1)
        sum += __shfl_xor(sum, off, 16);
      lrow[r] = lrow[r] * corr[r] + sum;
    }
#pragma unroll
    for (int t = 0; t < 8; ++t)
#pragma unroll
      for (int r = 0; r < 8; ++r) acc[t][r] *= corr[r];

    // stage P through LDS to convert C-layout -> A-layout (wave-internal, in-order)
#pragma unroll
    for (int r = 0; r < 8; ++r)
      pp[(r + 8 * half) * 16 + l16] = s[r];
    __builtin_amdgcn_wave_barrier();

    // acc += P @ V : V = h2[j0..j0+15][:]
#pragma unroll
    for (int kk = 0; kk < 4; ++kk) {
      int k0 = kk * 4 + 2 * half;
      v2f pa;
      pa.x = pp[l16 * 16 + k0];
      pa.y = pp[l16 * 16 + k0 + 1];
      const float* vr0 = h2 + (long)(j0 + k0) * D;
      const float* vr1 = h2 + (long)(j0 + k0 + 1) * D;
#pragma unroll
      for (int t = 0; t < 8; ++t) {
        v2f b;
        b.x = vr0[t * 16 + l16];
        b.y = vr1[t * 16 + l16];
        acc[t] = wmma_f32(pa, b, acc[t]);
      }
    }
    __builtin_amdgcn_wave_barrier();
  }

#pragma unroll
  for (int r = 0; r < 8; ++r) {
    float inv = 1.0f / lrow[r];
    int row = m0 + r + 8 * half;
#pragma unroll
    for (int t = 0; t < 8; ++t)
      ctx[(long)row * D + t * 16 + l16] = acc[t][r] * inv;
  }
}

// ---------------------------------------------------------------------------
// Per-subject pooling: aggr[b] = [max over h1 rows | mean over ctx rows]
// ---------------------------------------------------------------------------
__global__ void pool_kernel(const float* __restrict__ h1, const float* __restrict__ ctx,
                            float* __restrict__ aggr) {
  int b = blockIdx.x;
  int t = threadIdx.x;           // 0..255
  int d = t & 127;
  if (t < 128) {
    float m = -INFINITY;
    for (int i = 0; i < NPS; ++i)
      m = fmaxf(m, h1[(long)(b * NPS + i) * D + d]);
    aggr[b * 256 + d] = m;
  } else {
    float s = 0.f;
    for (int i = 0; i < NPS; ++i)
      s += ctx[(long)(b * NPS + i) * D + d];
    aggr[b * 256 + 128 + d] = s * (1.0f / (float)NPS);
  }
}

// ---------------------------------------------------------------------------
// MLP head with SiLU + training-mode BatchNorm (biased stats over 8 rows)
// ---------------------------------------------------------------------------
__global__ void head_kernel(const float* __restrict__ aggr,
                            const float* __restrict__ fc1_w, const float* __restrict__ fc1_b,
                            const float* __restrict__ fc2_w, const float* __restrict__ fc2_b,
                            const float* __restrict__ fc3_w, const float* __restrict__ fc3_b,
                            const float* __restrict__ bn1_g, const float* __restrict__ bn1_b,
                            const float* __restrict__ bn2_g, const float* __restrict__ bn2_b,
                            float* __restrict__ out) {
  __shared__ float s_ag[NSUBJ * 256];
  __shared__ float s_o1[NSUBJ * 512];
  __shared__ float s_o2[NSUBJ * 32];
  int t = threadIdx.x;  // 0..511
  for (int i = t; i < NSUBJ * 256; i += 512) s_ag[i] = aggr[i];
  __syncthreads();

  {   // fc1 column t, all 8 rows; SiLU; BN over the 8 rows held locally
    float z[NSUBJ];
#pragma unroll
    for (int b = 0; b < NSUBJ; ++b) {
      float acc = fc1_b[t];
      for (int k = 0; k < 256; ++k) acc += s_ag[b * 256 + k] * fc1_w[k * 512 + t];
      z[b] = acc / (1.f + expf(-acc));
    }
    float m = 0.f;
#pragma unroll
    for (int b = 0; b < NSUBJ; ++b) m += z[b];
    m *= 0.125f;
    float v = 0.f;
#pragma unroll
    for (int b = 0; b < NSUBJ; ++b) { float dd = z[b] - m; v += dd * dd; }
    v *= 0.125f;
    float isd = rsqrtf(v + 1e-5f);
    float g = bn1_g[t], bb = bn1_b[t];
#pragma unroll
    for (int b = 0; b < NSUBJ; ++b) s_o1[b * 512 + t] = (z[b] - m) * isd * g + bb;
  }
  __syncthreads();

  if (t < 32) {
    float z[NSUBJ];
#pragma unroll
    for (int b = 0; b < NSUBJ; ++b) {
      float acc = fc2_b[t];
      for (int k = 0; k < 512; ++k) acc += s_o1[b * 512 + k] * fc2_w[k * 32 + t];
      z[b] = acc / (1.f + expf(-acc));
    }
    float m = 0.f;
#pragma unroll
    for (int b = 0; b < NSUBJ; ++b) m += z[b];
    m *= 0.125f;
    float v = 0.f;
#pragma unroll
    for (int b = 0; b < NSUBJ; ++b) { float dd = z[b] - m; v += dd * dd; }
    v *= 0.125f;
    float isd = rsqrtf(v + 1e-5f);
    float g = bn2_g[t], bb = bn2_b[t];
#pragma unroll
    for (int b = 0; b < NSUBJ; ++b) s_o2[b * 32 + t] = (z[b] - m) * isd * g + bb;
  }
  __syncthreads();

  if (t < NSUBJ) {
    float l0 = fc3_b[0], l1 = fc3_b[1];
    for (int k = 0; k < 32; ++k) {
      float v = s_o2[t * 32 + k];
      l0 += v * fc3_w[k * 2 + 0];
      l1 += v * fc3_w[k * 2 + 1];
    }
    float mx = fmaxf(l0, l1);
    float e0 = expf(l0 - mx), e1 = expf(l1 - mx);
    float inv = 1.f / (e0 + e1);
    out[t * 2 + 0] = e0 * inv;
    out[t * 2 + 1] = e1 * inv;
  }
}

// ---------------------------------------------------------------------------
extern "C" void kernel_launch(void* const* d_in, const int* in_sizes, int n_in,
                              void* d_out, int out_size, void* d_ws, size_t ws_size,
                              hipStream_t stream) {
  const float* x     = (const float*)d_in[0];
  const float* W1    = (const float*)d_in[1];
  const float* b1    = (const float*)d_in[2];
  const float* W2    = (const float*)d_in[3];
  const float* b2    = (const float*)d_in[4];
  const float* pe    = (const float*)d_in[5];
  const float* fc1_w = (const float*)d_in[6];
  const float* fc1_b = (const float*)d_in[7];
  const float* fc2_w = (const float*)d_in[8];
  const float* fc2_b = (const float*)d_in[9];
  const float* fc3_w = (const float*)d_in[10];
  const float* fc3_b = (const float*)d_in[11];
  const float* bn1_g = (const float*)d_in[12];
  const float* bn1_b = (const float*)d_in[13];
  const float* bn2_g = (const float*)d_in[14];
  const float* bn2_b = (const float*)d_in[15];
  const int* edge_index = (const int*)d_in[16];
  const int* src = edge_index;                 // first E entries of [2,E]
  float* out = (float*)d_out;
  (void)in_sizes; (void)n_in; (void)out_size; (void)ws_size;

  float* ws = (float*)d_ws;
  size_t off = 0;
  float* agg1 = ws + off; off += (size_t)N_NODES * 84;
  float* h1   = ws + off; off += (size_t)N_NODES * D;
  float* agg2 = ws + off; off += (size_t)N_NODES * D;
  float* h2   = ws + off; off += (size_t)N_NODES * D;
  float* ctx  = ws + off; off += (size_t)N_NODES * D;
  float* aggr = ws + off; off += (size_t)NSUBJ * 256;

  dim3 gemm_grid(N_NODES / 16, D / 16);

  gin_agg_kernel<<<N_NODES, 128, 0, stream>>>(x, src, agg1, 84);
  wmma_gemm_kernel<<<gemm_grid, 32, 0, stream>>>(agg1, W1, b1, nullptr, h1, 84);
  gin_agg_kernel<<<N_NODES, 128, 0, stream>>>(h1, src, agg2, D);
  wmma_gemm_kernel<<<gemm_grid, 32, 0, stream>>>(agg2, W2, b2, pe, h2, D);
  flash_attn_kernel<<<N_NODES / 64, 128, 0, stream>>>(h2, ctx);
  pool_kernel<<<NSUBJ, 256, 0, stream>>>(h1, ctx, aggr);
  head_kernel<<<1, 512, 0, stream>>>(aggr, fc1_w, fc1_b, fc2_w, fc2_b, fc3_w, fc3_b,
                                     bn1_g, bn1_b, bn2_g, bn2_b, out);
}